// MinimalRNNCell_4672924418288
// MI455X (gfx1250) — compile-verified
//
#include <hip/hip_runtime.h>

// MinimalRNN linear recurrence on gfx1250 via chunked parallel scan + f32 WMMA.
//   s_{t+1} = x_t*KT + s_t*AT ; y_t = s_{t+1}*CyT
// Phase A: P = AT^64 (repeated squaring, 1 WG)
// Phase 1: per (batch-group, time-chunk) run recurrence from 0 -> chunk summary b_c
// Phase 2: serial scan over chunks: s_{c+1} = b_c + s_c*P  (depth 64)
// Phase 3: re-run chunks from true init states, emit y.
// All matmuls use V_WMMA_F32_16X16X4_F32 (native f32 tensor path; keeps
// reference precision over the 4096-step recurrence).

#define B_TOT 64
#define T_TOT 4096
#define NIN   64
#define U     128
#define NOUT  64
#define BC    16                 // batch rows per group (= WMMA M)
#define NG    (B_TOT / BC)       // 4 batch groups
#define LCH   64                 // timesteps per chunk
#define NCH   (T_TOT / LCH)      // 64 chunks
#define SROW  132                // padded LDS row stride (floats) -> conflict-free

typedef __attribute__((ext_vector_type(2))) float v2f;
typedef __attribute__((ext_vector_type(8))) float v8f;

__device__ __forceinline__ v8f wmma4(v2f a, v2f b, v8f c) {
  // D(16x16,f32) = A(16x4,f32) x B(4x16,f32) + C
  return __builtin_amdgcn_wmma_f32_16x16x4_f32(false, a, false, b,
                                               (short)0, c, false, false);
}

// ---------------------------------------------------------------- P = AT^64
__global__ __launch_bounds__(256) void k_matpow(const float* __restrict__ AT,
                                                float* __restrict__ P) {
  __shared__ float buf[U * U];   // 64 KB, in-place squaring (outputs staged in VGPRs)
  const int tid = threadIdx.x;
  const int w = tid >> 5, lane = tid & 31;
  const int h = lane >> 4, n = lane & 15, m = lane & 15;

  for (int i = tid; i < U * U; i += 256) buf[i] = AT[i];
  __syncthreads();

  for (int iter = 0; iter < 6; ++iter) {   // A^2,4,8,16,32,64
    v8f acc[8];
    #pragma unroll
    for (int i = 0; i < 8; ++i) {
      const int tile = w * 8 + i;
      const int mt = (tile >> 3) * 16, nt = (tile & 7) * 16;
      v8f c = {};
      #pragma unroll
      for (int kc = 0; kc < 32; ++kc) {
        const int k0 = kc * 4 + 2 * h;
        v2f a = *(const v2f*)&buf[(mt + m) * U + k0];
        v2f b;
        b.x = buf[k0 * U + nt + n];
        b.y = buf[(k0 + 1) * U + nt + n];
        c = wmma4(a, b, c);
      }
      acc[i] = c;
    }
    __syncthreads();
    #pragma unroll
    for (int i = 0; i < 8; ++i) {
      const int tile = w * 8 + i;
      const int mt = (tile >> 3) * 16, nt = (tile & 7) * 16;
      #pragma unroll
      for (int v = 0; v < 8; ++v) buf[(mt + v + 8 * h) * U + nt + n] = acc[i][v];
    }
    __syncthreads();
  }
  for (int i = tid; i < U * U; i += 256) P[i] = buf[i];
}

// -------------------------------------------- Phase 1: chunk summaries b_c
__global__ __launch_bounds__(256) void k_chunk_sum(const float* __restrict__ x,
                                                   const float* __restrict__ AT,
                                                   const float* __restrict__ KT,
                                                   float* __restrict__ bst) {
  __shared__ float sbuf[BC * SROW];
  const int c = blockIdx.x, g = blockIdx.y;
  const int tid = threadIdx.x, w = tid >> 5, lane = tid & 31;
  const int h = lane >> 4, n = lane & 15, m = lane & 15;

  // resident B-fragments for this wave's 16-wide unit slice
  v2f KTf[16], ATf[32];
  #pragma unroll
  for (int kc = 0; kc < 16; ++kc) {
    const int k0 = kc * 4 + 2 * h;
    KTf[kc].x = KT[k0 * U + w * 16 + n];
    KTf[kc].y = KT[(k0 + 1) * U + w * 16 + n];
  }
  #pragma unroll
  for (int kc = 0; kc < 32; ++kc) {
    const int k0 = kc * 4 + 2 * h;
    ATf[kc].x = AT[k0 * U + w * 16 + n];
    ATf[kc].y = AT[(k0 + 1) * U + w * 16 + n];
  }
  for (int i = tid; i < BC * SROW; i += 256) sbuf[i] = 0.0f;
  __syncthreads();

  const long tbase = (long)(g * BC + m) * T_TOT;
  for (int tl = 0; tl < LCH; ++tl) {
    const int t = c * LCH + tl;
    v8f acc = {};
    const float* xrow = x + (tbase + t) * NIN;
    #pragma unroll
    for (int kc = 0; kc < 16; ++kc)          // ky_t = x_t * KT
      acc = wmma4(*(const v2f*)&xrow[kc * 4 + 2 * h], KTf[kc], acc);
    #pragma unroll
    for (int kc = 0; kc < 32; ++kc)          // + s_t * AT
      acc = wmma4(*(const v2f*)&sbuf[m * SROW + kc * 4 + 2 * h], ATf[kc], acc);
    __syncthreads();
    #pragma unroll
    for (int v = 0; v < 8; ++v) sbuf[(v + 8 * h) * SROW + w * 16 + n] = acc[v];
    __syncthreads();
  }
  float* o = bst + (long)(g * NCH + c) * BC * U;
  #pragma unroll
  for (int v = 0; v < 8; ++v)
    o[(v + 8 * h) * U + w * 16 + n] = sbuf[(v + 8 * h) * SROW + w * 16 + n];
}

// ---------------------- Phase 2: scan over chunks, bst becomes init states
__global__ __launch_bounds__(256) void k_scan(const float* __restrict__ P,
                                              float* __restrict__ bst) {
  __shared__ float sbuf[BC * SROW];
  const int g = blockIdx.x;
  const int tid = threadIdx.x, w = tid >> 5, lane = tid & 31;
  const int h = lane >> 4, n = lane & 15, m = lane & 15;

  v2f Pf[32];
  #pragma unroll
  for (int kc = 0; kc < 32; ++kc) {
    const int k0 = kc * 4 + 2 * h;
    Pf[kc].x = P[k0 * U + w * 16 + n];
    Pf[kc].y = P[(k0 + 1) * U + w * 16 + n];
  }
  for (int i = tid; i < BC * SROW; i += 256) sbuf[i] = 0.0f;
  __syncthreads();

  for (int c = 0; c < NCH; ++c) {
    float* slot = bst + (long)(g * NCH + c) * BC * U;
    v8f acc;
    #pragma unroll
    for (int v = 0; v < 8; ++v) {
      const int idx = (v + 8 * h) * U + w * 16 + n;
      acc[v] = slot[idx];                                   // read b_c
      slot[idx] = sbuf[(v + 8 * h) * SROW + w * 16 + n];    // overwrite with s_c
    }
    #pragma unroll
    for (int kc = 0; kc < 32; ++kc)                          // s_{c+1} = b_c + s_c*P
      acc = wmma4(*(const v2f*)&sbuf[m * SROW + kc * 4 + 2 * h], Pf[kc], acc);
    __syncthreads();
    #pragma unroll
    for (int v = 0; v < 8; ++v) sbuf[(v + 8 * h) * SROW + w * 16 + n] = acc[v];
    __syncthreads();
  }
}

// ----------------------------- Phase 3: replay chunks with true init, emit y
__global__ __launch_bounds__(256) void k_chunk_out(const float* __restrict__ x,
                                                   const float* __restrict__ AT,
                                                   const float* __restrict__ KT,
                                                   const float* __restrict__ CyT,
                                                   const float* __restrict__ bst,
                                                   float* __restrict__ out) {
  __shared__ float sbuf[BC * SROW];
  const int c = blockIdx.x, g = blockIdx.y;
  const int tid = threadIdx.x, w = tid >> 5, lane = tid & 31;
  const int h = lane >> 4, n = lane & 15, m = lane & 15;

  v2f KTf[16], ATf[32], Cyf[32];
  #pragma unroll
  for (int kc = 0; kc < 16; ++kc) {
    const int k0 = kc * 4 + 2 * h;
    KTf[kc].x = KT[k0 * U + w * 16 + n];
    KTf[kc].y = KT[(k0 + 1) * U + w * 16 + n];
  }
  #pragma unroll
  for (int kc = 0; kc < 32; ++kc) {
    const int k0 = kc * 4 + 2 * h;
    ATf[kc].x = AT[k0 * U + w * 16 + n];
    ATf[kc].y = AT[(k0 + 1) * U + w * 16 + n];
  }
  if (w < 4) {                               // waves 0-3 own the 16x64 y tile
    #pragma unroll
    for (int kc = 0; kc < 32; ++kc) {
      const int k0 = kc * 4 + 2 * h;
      Cyf[kc].x = CyT[k0 * NOUT + w * 16 + n];
      Cyf[kc].y = CyT[(k0 + 1) * NOUT + w * 16 + n];
    }
  }

  const float* s0 = bst + (long)(g * NCH + c) * BC * U;
  for (int i = tid; i < BC * U; i += 256)
    sbuf[(i >> 7) * SROW + (i & 127)] = s0[i];
  __syncthreads();

  const long tbase = (long)(g * BC + m) * T_TOT;
  for (int tl = 0; tl < LCH; ++tl) {
    const int t = c * LCH + tl;
    v8f acc = {};
    const float* xrow = x + (tbase + t) * NIN;
    #pragma unroll
    for (int kc = 0; kc < 16; ++kc)
      acc = wmma4(*(const v2f*)&xrow[kc * 4 + 2 * h], KTf[kc], acc);
    #pragma unroll
    for (int kc = 0; kc < 32; ++kc)
      acc = wmma4(*(const v2f*)&sbuf[m * SROW + kc * 4 + 2 * h], ATf[kc], acc);
    __syncthreads();
    #pragma unroll
    for (int v = 0; v < 8; ++v) sbuf[(v + 8 * h) * SROW + w * 16 + n] = acc[v];
    __syncthreads();
    if (w < 4) {                             // y_t = s_{t+1} * CyT
      v8f y = {};
      #pragma unroll
      for (int kc = 0; kc < 32; ++kc)
        y = wmma4(*(const v2f*)&sbuf[m * SROW + kc * 4 + 2 * h], Cyf[kc], y);
      #pragma unroll
      for (int v = 0; v < 8; ++v)
        out[((long)(g * BC + v + 8 * h) * T_TOT + t) * NOUT + w * 16 + n] = y[v];
    }
  }
}

extern "C" void kernel_launch(void* const* d_in, const int* in_sizes, int n_in,
                              void* d_out, int out_size, void* d_ws, size_t ws_size,
                              hipStream_t stream) {
  const float* x   = (const float*)d_in[0];
  const float* AT  = (const float*)d_in[1];
  const float* KT  = (const float*)d_in[2];
  const float* CyT = (const float*)d_in[3];
  float* out = (float*)d_out;
  // scratch: P (64 KB) | chunk states (NG*NCH*BC*U f32 = 2 MB)
  float* P   = (float*)d_ws;
  float* bst = P + U * U;

  k_matpow  <<<1,                 256, 0, stream>>>(AT, P);
  k_chunk_sum<<<dim3(NCH, NG),    256, 0, stream>>>(x, AT, KT, bst);
  k_scan    <<<NG,                256, 0, stream>>>(P, bst);
  k_chunk_out<<<dim3(NCH, NG),    256, 0, stream>>>(x, AT, KT, CyT, bst, out);
}